// ClusterInversionLoss_11433202942578
// MI455X (gfx1250) — compile-verified
//
#include <hip/hip_runtime.h>
#include <hip/hip_bf16.h>
#include <math.h>

typedef __attribute__((ext_vector_type(2))) float v2f;
typedef __attribute__((ext_vector_type(8))) float v8f;

#define NUM_C 5
#define LOSS_EPS 1e-8f
#define PAIR_BLOCKS 2048
#define PAIR_BS 256

// ---------------------------------------------------------------------------
// Kernel 1: s[i] = sum_k k * softmax(inputs[i,:])_k   (N x 5, sequential reads)
// ---------------------------------------------------------------------------
__device__ __forceinline__ float ordinal_score(const float* __restrict__ row) {
    float x0 = row[0], x1 = row[1], x2 = row[2], x3 = row[3], x4 = row[4];
    float m = fmaxf(fmaxf(fmaxf(x0, x1), fmaxf(x2, x3)), x4);
    float e0 = expf(x0 - m), e1 = expf(x1 - m), e2 = expf(x2 - m),
          e3 = expf(x3 - m), e4 = expf(x4 - m);
    float denom = e0 + e1 + e2 + e3 + e4;
    float numer = e1 + 2.0f * e2 + 3.0f * e3 + 4.0f * e4;
    return numer / denom;
}

__global__ void score_kernel(const float* __restrict__ inputs,
                             float* __restrict__ s, int N) {
    int i = blockIdx.x * blockDim.x + threadIdx.x;
    if (i >= N) return;
    s[i] = ordinal_score(inputs + (size_t)i * NUM_C);
}

// ---------------------------------------------------------------------------
// Kernel 2: per-pair loss, block-level deterministic reduction to partials.
//           partials[0..nb)   = per-block loss sums
//           partials[nb..2nb) = per-block weight sums
// ---------------------------------------------------------------------------
__global__ void pair_kernel(const int* __restrict__ pair_i,
                            const int* __restrict__ pair_j,
                            const int* __restrict__ targets,
                            const float* __restrict__ sweight,
                            const float* __restrict__ s,        // may be null
                            const float* __restrict__ inputs,   // fallback path
                            int npairs,
                            float* __restrict__ partials, int nb) {
    __shared__ float red_l[PAIR_BS];
    __shared__ float red_w[PAIR_BS];

    const int tid    = threadIdx.x;
    const int stride = gridDim.x * blockDim.x;
    float accL = 0.0f, accW = 0.0f;

    for (int idx = blockIdx.x * blockDim.x + tid; idx < npairs; idx += stride) {
        // prefetch next grid-stride pass of the streaming index arrays
        if (idx + stride < npairs) {
            __builtin_prefetch(pair_i + idx + stride, 0, 0);
            __builtin_prefetch(pair_j + idx + stride, 0, 0);
        }
        int pi = pair_i[idx];
        int pj = pair_j[idx];
        int yi = targets[pi];
        int yj = targets[pj];
        int d  = yi - yj;
        if (d != 0) {                       // ties (incl. i==j) contribute 0
            float si = s ? s[pi] : ordinal_score(inputs + (size_t)pi * NUM_C);
            float sj = s ? s[pj] : ordinal_score(inputs + (size_t)pj * NUM_C);
            float wp = 0.5f * (sweight[pi] + sweight[pj]);
            float sign  = (d > 0) ? 1.0f : -1.0f;
            float delta = sign * (si - sj);
            float z  = -delta;              // MARGIN == 0
            // stable softplus: log1p(exp(z)) = max(z,0) + log1p(exp(-|z|))
            float sp = fmaxf(z, 0.0f) + log1pf(expf(-fabsf(z)));
            float dist = fabsf((float)d);
            accL += sp * dist * wp;
            accW += wp;
        }
    }

    red_l[tid] = accL;
    red_w[tid] = accW;
    __syncthreads();
    for (int off = PAIR_BS / 2; off > 0; off >>= 1) {
        if (tid < off) {
            red_l[tid] += red_l[tid + off];
            red_w[tid] += red_w[tid + off];
        }
        __syncthreads();
    }
    if (tid == 0) {
        partials[blockIdx.x]      = red_l[0];
        partials[nb + blockIdx.x] = red_w[0];
    }
}

// ---------------------------------------------------------------------------
// Kernel 3: single-wave final reduction via V_WMMA_F32_16X16X4_F32.
//   A vgpr0 = per-lane loss partial   -> A[m, k=0|2]
//   A vgpr1 = per-lane weight partial -> A[m, k=1|3]
//   B col0 = [1,0,1,0]^T, col1 = [0,1,0,1]^T
//   => D[m,0] = loss[m]+loss[m+16], D[m,1] = weight[m]+weight[m+16]
//   Lane t sums its 8 D VGPRs; totals = t(0)+t(16) and t(1)+t(17).
// ---------------------------------------------------------------------------
__global__ void final_kernel(const float* __restrict__ partials, int nb,
                             float* __restrict__ out) {
    int lane = threadIdx.x;                // 32 threads = 1 full wave
    float L = 0.0f, W = 0.0f;
    for (int k = lane; k < nb; k += 32) {  // nb multiple of 32: no divergence
        L += partials[k];
        W += partials[nb + k];
    }
    int n = lane & 15;
    v2f a; a.x = L; a.y = W;
    v2f b; b.x = (n == 0) ? 1.0f : 0.0f;   // B[k=0|2, n]
           b.y = (n == 1) ? 1.0f : 0.0f;   // B[k=1|3, n]
    v8f c = {};
    v8f dm = __builtin_amdgcn_wmma_f32_16x16x4_f32(
        false, a, false, b, (short)0, c, false, false);
    float t = dm[0] + dm[1] + dm[2] + dm[3] + dm[4] + dm[5] + dm[6] + dm[7];
    float Ltot = __shfl(t, 0, 32)  + __shfl(t, 16, 32);
    float Wtot = __shfl(t, 1, 32)  + __shfl(t, 17, 32);
    if (lane == 0) out[0] = Ltot / (Wtot + LOSS_EPS);
}

// ---------------------------------------------------------------------------
extern "C" void kernel_launch(void* const* d_in, const int* in_sizes, int n_in,
                              void* d_out, int out_size, void* d_ws, size_t ws_size,
                              hipStream_t stream) {
    const float* inputs  = (const float*)d_in[0];
    const int*   targets = (const int*)d_in[1];
    // d_in[2] = cluster_ids: unused by the reference computation
    const float* sweight = (const float*)d_in[3];
    const int*   pair_i  = (const int*)d_in[4];
    const int*   pair_j  = (const int*)d_in[5];

    const int N      = in_sizes[1];
    const int npairs = in_sizes[4];
    float* out = (float*)d_out;
    float* ws  = (float*)d_ws;

    const size_t need = (size_t)N * sizeof(float) + (size_t)2 * PAIR_BLOCKS * sizeof(float);

    float* s = nullptr;
    float* partials;
    if (ws_size >= need) {
        s = ws;
        partials = ws + N;
        int blocks = (N + PAIR_BS - 1) / PAIR_BS;
        score_kernel<<<blocks, PAIR_BS, 0, stream>>>(inputs, s, N);
    } else {
        partials = ws;   // fallback: recompute scores inline at gather time
    }

    pair_kernel<<<PAIR_BLOCKS, PAIR_BS, 0, stream>>>(
        pair_i, pair_j, targets, sweight, s, inputs, npairs, partials, PAIR_BLOCKS);

    final_kernel<<<1, 32, 0, stream>>>(partials, PAIR_BLOCKS, out);
}